// Agent_52991306498170
// MI455X (gfx1250) — compile-verified
//
#include <hip/hip_runtime.h>

typedef __attribute__((ext_vector_type(2))) float v2f;
typedef __attribute__((ext_vector_type(8))) float v8f;

namespace {

constexpr int kH = 8;    // hidden dim
constexpr int kWP = 16;  // padded B columns in LDS (cols 8..15 zero)

// deg starts at 1.0 (self-loop); also zero the mean accumulator.
__global__ void init_kernel(float* __restrict__ deg, float* __restrict__ macc, int n) {
  int i = blockIdx.x * blockDim.x + threadIdx.x;
  if (i < n) deg[i] = 1.0f;
  if (i < kH) macc[i] = 0.0f;
}

__global__ void deg_kernel(const long long* __restrict__ ei, int E, float* __restrict__ deg) {
  int e = blockIdx.x * blockDim.x + threadIdx.x;
  if (e >= E) return;
  int c = (int)ei[(size_t)E + (size_t)e];  // target node
  atomicAdd(&deg[c], 1.0f);
}

__global__ void rsqrt_kernel(float* __restrict__ deg, int n) {
  int i = blockIdx.x * blockDim.x + threadIdx.x;
  if (i < n) deg[i] = 1.0f / sqrtf(deg[i]);  // deg >= 1 always (self-loop)
}

// Initialize the aggregation buffer with the self-loop message: G = T * dis^2.
__global__ void selfmsg_kernel(float* __restrict__ G, const float* __restrict__ T,
                               const float* __restrict__ dis, int n) {
  int i = blockIdx.x * blockDim.x + threadIdx.x;
  if (i >= n) return;
  const float sn = dis[i] * dis[i];
  const float4* t4 = (const float4*)(T + (size_t)i * kH);
  float4* g4 = (float4*)(G + (size_t)i * kH);
  float4 t0 = t4[0], t1 = t4[1];
  t0.x *= sn; t0.y *= sn; t0.z *= sn; t0.w *= sn;
  t1.x *= sn; t1.y *= sn; t1.z *= sn; t1.w *= sn;
  g4[0] = t0;
  g4[1] = t1;
}

// out(n x 8) = in(n x K) @ W(K x 8) via V_WMMA_F32_16X16X4_F32.
// One wave computes a 16-row tile. Branch-free inner loop AND branch-free store:
//  - A rows clamped to n-1 (garbage D rows land in out's padding, never read)
//  - W held in LDS padded to 16 columns (cols 8..15 = 0) so B loads are unconditional
//  - out must have room for ceil(n/16)*16 rows (workspace buffers are padded)
template <int K>
__global__ void gemm_wmma_kernel(const float* __restrict__ in, const float* __restrict__ W,
                                 float* __restrict__ out, int n) {
  __shared__ float sW[K * kWP];
  for (int t = threadIdx.x; t < K * kWP; t += blockDim.x) sW[t] = 0.0f;
  __syncthreads();
  for (int t = threadIdx.x; t < K * kH; t += blockDim.x) {
    sW[(t >> 3) * kWP + (t & 7)] = W[t];
  }
  __syncthreads();

  const int lane = threadIdx.x & 31;
  const int wid = threadIdx.x >> 5;
  const int tile = blockIdx.x * (blockDim.x >> 5) + wid;
  const int rowBase = tile * 16;
  if (rowBase >= n) return;  // wave-uniform exit: active waves keep full EXEC for WMMA

  const int g = lane >> 4;   // lane group: 0 = lanes 0-15, 1 = lanes 16-31
  const int q = lane & 15;   // A: row offset (M); B/D: column (N)
  const int m = min(rowBase + q, n - 1);  // clamped: branch-free A loads

  const float* __restrict__ ap = in + (size_t)m * K + 2 * g;
  const float* __restrict__ bp = sW + 2 * g * kWP + q;

  v8f c = {};
#pragma unroll 8
  for (int k = 0; k < K; k += 4) {
    // A 16x4 f32: lane holds {A[m][k+2g], A[m][k+2g+1]}  (8B-aligned)
    const float2 av = *(const float2*)(ap + k);
    v2f a;
    a.x = av.x;
    a.y = av.y;
    // B 4x16 f32: lane holds {B[k+2g][q], B[k+2g+1][q]}
    v2f b;
    b.x = bp[k * kWP];
    b.y = bp[k * kWP + kWP];
    c = __builtin_amdgcn_wmma_f32_16x16x4_f32(false, a, false, b, (short)0, c, false, false);
  }

  // D 16x16 f32: lane l, VGPR v holds D[v + 8*g][q].
  // Unconditional stores: rows [n, rowBase+16) fall in out's padding.
  if (q < kH) {
    float* __restrict__ op = out + (size_t)(rowBase + 8 * g) * kH + q;
#pragma unroll
    for (int v = 0; v < 8; ++v) {
      op[v * kH] = c[v];
    }
  }
}

// G[col] += T[row] * (dis[row]*dis[col]) over all directed edges.
__global__ void scatter_kernel(const long long* __restrict__ ei, int E,
                               const float* __restrict__ dis, const float* __restrict__ T,
                               float* __restrict__ G) {
  int e = blockIdx.x * blockDim.x + threadIdx.x;
  if (e >= E) return;
  const int r = (int)ei[e];
  const int c = (int)ei[(size_t)E + (size_t)e];
  const float nrm = dis[r] * dis[c];
  const float4* src = (const float4*)(T + (size_t)r * kH);
  const float4 m0 = src[0];
  const float4 m1 = src[1];
  float* dst = G + (size_t)c * kH;
  atomicAdd(dst + 0, m0.x * nrm);
  atomicAdd(dst + 1, m0.y * nrm);
  atomicAdd(dst + 2, m0.z * nrm);
  atomicAdd(dst + 3, m0.w * nrm);
  atomicAdd(dst + 4, m1.x * nrm);
  atomicAdd(dst + 5, m1.y * nrm);
  atomicAdd(dst + 6, m1.z * nrm);
  atomicAdd(dst + 7, m1.w * nrm);
}

// G = relu(G + b), in place (self-loop message already folded in by selfmsg_kernel).
__global__ void epilogue_kernel(float* __restrict__ G, const float* __restrict__ b, int n) {
  int i = blockIdx.x * blockDim.x + threadIdx.x;
  if (i >= n) return;
  float4* g4 = (float4*)(G + (size_t)i * kH);
  float4 g0 = g4[0], g1 = g4[1];
  g0.x = fmaxf(g0.x + b[0], 0.f);
  g0.y = fmaxf(g0.y + b[1], 0.f);
  g0.z = fmaxf(g0.z + b[2], 0.f);
  g0.w = fmaxf(g0.w + b[3], 0.f);
  g1.x = fmaxf(g1.x + b[4], 0.f);
  g1.y = fmaxf(g1.y + b[5], 0.f);
  g1.z = fmaxf(g1.z + b[6], 0.f);
  g1.w = fmaxf(g1.w + b[7], 0.f);
  g4[0] = g0;
  g4[1] = g1;
}

// proba[i] = h[i] . Wp + bp ; also block-reduce sum(h) into macc for the mean pool.
__global__ void final_kernel(const float* __restrict__ h, const float* __restrict__ Wp,
                             const float* __restrict__ bp, float* __restrict__ macc,
                             float* __restrict__ out, int n) {
  __shared__ float red[kH * 256];
  const int tid = threadIdx.x;
  const int i = blockIdx.x * blockDim.x + tid;
  float v[kH];
  if (i < n) {
    const float4* h4 = (const float4*)(h + (size_t)i * kH);
    const float4 a = h4[0], bb = h4[1];
    v[0] = a.x; v[1] = a.y; v[2] = a.z; v[3] = a.w;
    v[4] = bb.x; v[5] = bb.y; v[6] = bb.z; v[7] = bb.w;
    float s = bp[0];
#pragma unroll
    for (int f = 0; f < kH; ++f) s += v[f] * Wp[f];
    out[i] = s;
  } else {
#pragma unroll
    for (int f = 0; f < kH; ++f) v[f] = 0.f;
  }
#pragma unroll
  for (int f = 0; f < kH; ++f) red[f * 256 + tid] = v[f];
  __syncthreads();
  for (int stride = 128; stride > 0; stride >>= 1) {
    if (tid < stride) {
#pragma unroll
      for (int f = 0; f < kH; ++f) red[f * 256 + tid] += red[f * 256 + tid + stride];
    }
    __syncthreads();
  }
  if (tid < kH) atomicAdd(&macc[tid], red[tid * 256]);
}

__global__ void value_kernel(const float* __restrict__ macc, const float* __restrict__ Wv,
                             const float* __restrict__ bv, int n, float* __restrict__ out) {
  if (threadIdx.x == 0) {
    const float inv = 1.0f / (float)n;
    float s = bv[0];
#pragma unroll
    for (int f = 0; f < kH; ++f) s += macc[f] * inv * Wv[f];
    out[0] = s;
  }
}

}  // namespace

extern "C" void kernel_launch(void* const* d_in, const int* in_sizes, int n_in,
                              void* d_out, int out_size, void* d_ws, size_t ws_size,
                              hipStream_t stream) {
  (void)n_in; (void)out_size; (void)ws_size;
  const float* x      = (const float*)d_in[0];
  const long long* ei = (const long long*)d_in[1];  // int64 [2, E]
  const float* W1 = (const float*)d_in[2];
  const float* b1 = (const float*)d_in[3];
  const float* W2 = (const float*)d_in[4];
  const float* b2 = (const float*)d_in[5];
  const float* W3 = (const float*)d_in[6];
  const float* b3 = (const float*)d_in[7];
  const float* Wp = (const float*)d_in[8];
  const float* bp = (const float*)d_in[9];
  const float* Wv = (const float*)d_in[10];
  const float* bv = (const float*)d_in[11];

  const int N = in_sizes[0] / 128;
  const int E = in_sizes[1] / 2;

  // Workspace layout (floats): deg/dis | buf0 (T) | buf1 | buf2 | mean acc  ~= 10 MB
  // Node buffers padded to a multiple of 16 rows so WMMA tile stores need no guard.
  float* wsf = (float*)d_ws;
  const size_t nPad = ((size_t)N + 15) & ~(size_t)15;
  float* deg  = wsf;
  float* buf0 = deg + nPad;
  float* buf1 = buf0 + nPad * kH;
  float* buf2 = buf1 + nPad * kH;
  float* macc = buf2 + nPad * kH;
  float* proba = (float*)d_out;

  const int TB = 256;
  auto cdiv = [](int a, int b) { return (a + b - 1) / b; };
  const int gemmBlocks = cdiv(N, 128);  // 8 waves/block x 16 rows/wave

  init_kernel<<<cdiv(N, TB), TB, 0, stream>>>(deg, macc, N);
  deg_kernel<<<cdiv(E, TB), TB, 0, stream>>>(ei, E, deg);
  rsqrt_kernel<<<cdiv(N, TB), TB, 0, stream>>>(deg, N);  // deg -> dis in place

  // Layer 1: x(N x 128) @ W1 -> buf0; aggregate into buf1; h1 = buf1
  gemm_wmma_kernel<128><<<gemmBlocks, TB, 0, stream>>>(x, W1, buf0, N);
  selfmsg_kernel<<<cdiv(N, TB), TB, 0, stream>>>(buf1, buf0, deg, N);
  scatter_kernel<<<cdiv(E, TB), TB, 0, stream>>>(ei, E, deg, buf0, buf1);
  epilogue_kernel<<<cdiv(N, TB), TB, 0, stream>>>(buf1, b1, N);

  // Layer 2: h1 @ W2 -> buf0; aggregate into buf2; h2 = buf2
  gemm_wmma_kernel<8><<<gemmBlocks, TB, 0, stream>>>(buf1, W2, buf0, N);
  selfmsg_kernel<<<cdiv(N, TB), TB, 0, stream>>>(buf2, buf0, deg, N);
  scatter_kernel<<<cdiv(E, TB), TB, 0, stream>>>(ei, E, deg, buf0, buf2);
  epilogue_kernel<<<cdiv(N, TB), TB, 0, stream>>>(buf2, b2, N);

  // Layer 3: h2 @ W3 -> buf0; aggregate into buf1; h3 = buf1
  gemm_wmma_kernel<8><<<gemmBlocks, TB, 0, stream>>>(buf2, W3, buf0, N);
  selfmsg_kernel<<<cdiv(N, TB), TB, 0, stream>>>(buf1, buf0, deg, N);
  scatter_kernel<<<cdiv(E, TB), TB, 0, stream>>>(ei, E, deg, buf0, buf1);
  epilogue_kernel<<<cdiv(N, TB), TB, 0, stream>>>(buf1, b3, N);

  // Heads: proba per node + mean-pooled value
  final_kernel<<<cdiv(N, TB), TB, 0, stream>>>(buf1, Wp, bp, macc, proba, N);
  value_kernel<<<1, 32, 0, stream>>>(macc, Wv, bv, N, proba + N);
}